// Navi_Diego_69827578298542
// MI455X (gfx1250) — compile-verified
//
#include <hip/hip_runtime.h>

// Problem constants (from reference setup_inputs)
#define DIM   512
#define NREL  4

typedef __attribute__((ext_vector_type(2))) float v2f;
typedef __attribute__((ext_vector_type(8))) float v8f;

// ---------------------------------------------------------------------------
// Zero a float buffer (grid-stride)
// ---------------------------------------------------------------------------
__global__ void rgcn_zero_kernel(float* __restrict__ p, long n) {
    long i = (long)blockIdx.x * blockDim.x + threadIdx.x;
    long stride = (long)gridDim.x * blockDim.x;
    for (; i < n; i += stride) p[i] = 0.0f;
}

// ---------------------------------------------------------------------------
// Scatter: AX[s,:] += X[g,:]  ; deg[s] += 1 ; full_deg[s] += 1
// One block (128 threads) per edge; each thread moves a float4 (512 floats/row).
// X and AX are L2-resident (102 MB each, 192 MB L2) -> L2-bw bound atomics.
// ---------------------------------------------------------------------------
__global__ void rgcn_scatter_kernel(const float* __restrict__ X,
                                    const int* __restrict__ gather_idx,
                                    const int* __restrict__ scatter_idx,
                                    float* __restrict__ AX,
                                    float* __restrict__ deg,
                                    float* __restrict__ full_deg) {
    const int e = blockIdx.x;
    const int g = gather_idx[e];
    const int s = scatter_idx[e];
    const int t = threadIdx.x;                       // 0..127
    const float4 v = *(const float4*)(X + (size_t)g * DIM + t * 4);
    float* dst = AX + (size_t)s * DIM + t * 4;
    atomicAdd(dst + 0, v.x);
    atomicAdd(dst + 1, v.y);
    atomicAdd(dst + 2, v.z);
    atomicAdd(dst + 3, v.w);
    if (t == 0) {
        atomicAdd(deg + s, 1.0f);
        atomicAdd(full_deg + s, 1.0f);
    }
}

// ---------------------------------------------------------------------------
// GEMM + epilogue: out += AX @ W + deg[:,None] * bias
// Block = 256 threads (8 waves). Block tile: 16 rows x 512 cols.
// Each wave: 16x64 tile = 4 x v8f accumulators (4 independent WMMA chains).
// Matrix op: V_WMMA_F32_16X16X4_F32 (the CDNA5 fp32 WMMA shape).
//   A layout (32-bit, 16x4): lanes 0-15 -> M=lane, K={k0,k0+1};
//                            lanes 16-31 -> M=lane-16, K={k0+2,k0+3}
//   B layout (4x16): lanes 0-15 -> N=lane, K={k0,k0+1};
//                    lanes 16-31 -> N=lane-16, K={k0+2,k0+3}
//   C/D layout: VGPR v, lanes 0-15 -> M=v, N=lane; lanes 16-31 -> M=v+8.
// LDS staging is PAIR-SWIZZLED: Bs2[(k/2)*DIM + col] = {W[k][col], W[k+1][col]}
// so every B fragment is one aligned ds_load_b64 directly into the WMMA
// source register pair (no v_dual_mov shuffle tail, half the DS ops).
// ---------------------------------------------------------------------------
__global__ void __launch_bounds__(256)
rgcn_gemm_kernel(const float* __restrict__ AX,
                 const float* __restrict__ W,      // DIM x DIM row-major (K x N)
                 const float* __restrict__ bias,   // DIM
                 const float* __restrict__ deg,    // N nodes
                 float* __restrict__ out) {
    __shared__ float2 Bs2[8 * DIM];                 // 32 KB: 16-K slice, pair-major

    const int lane    = threadIdx.x & 31;
    const int wave    = threadIdx.x >> 5;           // 0..7
    const int l15     = lane & 15;
    const int halfSel = lane >> 4;                  // 0 or 1
    const int row0    = blockIdx.x * 16;
    const int colBase = wave * 64;

    v8f c0 = {}, c1 = {}, c2 = {}, c3 = {};

    const float* Arow = AX + (size_t)(row0 + l15) * DIM;

    for (int k0 = 0; k0 < DIM; k0 += 16) {
        // Cooperative pair-swizzled staging of W[k0:k0+16][0:512] into LDS.
        // idx = kp*512 + col ; reads W[k0+2kp][col], W[k0+2kp+1][col] (coalesced
        // across threads for fixed iteration), one ds_store_b64 each.
        #pragma unroll
        for (int i = 0; i < (8 * DIM) / 256; ++i) {
            const int idx = threadIdx.x + i * 256;
            const int kp  = idx >> 9;               // 0..7
            const int col = idx & (DIM - 1);
            const float* wp = W + (size_t)(k0 + 2 * kp) * DIM + col;
            float2 pr;
            pr.x = wp[0];
            pr.y = wp[DIM];
            Bs2[idx] = pr;
        }
        __syncthreads();

        #pragma unroll
        for (int kk = 0; kk < 16; kk += 4) {
            const int kb = kk + halfSel * 2;        // even
            const float2 af = *(const float2*)(Arow + k0 + kb);
            v2f a; a.x = af.x; a.y = af.y;

            const float2* bp = Bs2 + (kb >> 1) * DIM;  // pair row for this lane-half

            float2 bf; v2f b;
            bf = bp[colBase + 0 * 16 + l15];
            b.x = bf.x; b.y = bf.y;
            c0 = __builtin_amdgcn_wmma_f32_16x16x4_f32(false, a, false, b,
                                                       (short)0, c0, false, false);
            bf = bp[colBase + 1 * 16 + l15];
            b.x = bf.x; b.y = bf.y;
            c1 = __builtin_amdgcn_wmma_f32_16x16x4_f32(false, a, false, b,
                                                       (short)0, c1, false, false);
            bf = bp[colBase + 2 * 16 + l15];
            b.x = bf.x; b.y = bf.y;
            c2 = __builtin_amdgcn_wmma_f32_16x16x4_f32(false, a, false, b,
                                                       (short)0, c2, false, false);
            bf = bp[colBase + 3 * 16 + l15];
            b.x = bf.x; b.y = bf.y;
            c3 = __builtin_amdgcn_wmma_f32_16x16x4_f32(false, a, false, b,
                                                       (short)0, c3, false, false);
        }
        __syncthreads();
    }

    // Epilogue: out += C + deg[row] * bias[col]
    #pragma unroll
    for (int v = 0; v < 8; ++v) {
        const int row = row0 + v + halfSel * 8;
        const float dr = deg[row];
        float* orow = out + (size_t)row * DIM;
        {
            const int col = colBase + 0 * 16 + l15;
            orow[col] += c0[v] + dr * bias[col];
        }
        {
            const int col = colBase + 1 * 16 + l15;
            orow[col] += c1[v] + dr * bias[col];
        }
        {
            const int col = colBase + 2 * 16 + l15;
            orow[col] += c2[v] + dr * bias[col];
        }
        {
            const int col = colBase + 3 * 16 + l15;
            orow[col] += c3[v] + dr * bias[col];
        }
    }
}

// ---------------------------------------------------------------------------
// Final scale: out[i,:] *= 1/full_deg[i] (full_deg==0 treated as 1)
// One block (128 threads, float4) per node row.
// ---------------------------------------------------------------------------
__global__ void rgcn_finalize_kernel(float* __restrict__ out,
                                     const float* __restrict__ full_deg) {
    const int i = blockIdx.x;
    const float fd = full_deg[i];
    const float inv = 1.0f / (fd == 0.0f ? 1.0f : fd);
    float4* row = (float4*)(out + (size_t)i * DIM);
    float4 v = row[threadIdx.x];
    v.x *= inv; v.y *= inv; v.z *= inv; v.w *= inv;
    row[threadIdx.x] = v;
}

// ---------------------------------------------------------------------------
// Driver
// ---------------------------------------------------------------------------
extern "C" void kernel_launch(void* const* d_in, const int* in_sizes, int n_in,
                              void* d_out, int out_size, void* d_ws, size_t ws_size,
                              hipStream_t stream) {
    const float* features = (const float*)d_in[0];   // N x D
    const float* w        = (const float*)d_in[1];   // R x D x D
    const float* bias     = (const float*)d_in[2];   // R x D
    const float* w_t      = (const float*)d_in[3];   // R x D x D
    const float* bias_t   = (const float*)d_in[4];   // R x D
    const int*   edge_src = (const int*)d_in[5];     // R x E
    const int*   edge_dst = (const int*)d_in[6];     // R x E

    const int N = in_sizes[0] / DIM;                 // 50000
    const int E = in_sizes[5] / NREL;                // 400000

    float* out      = (float*)d_out;                 // N x D
    float* AX       = (float*)d_ws;                  // N x D
    float* deg      = AX + (size_t)N * DIM;          // N
    float* full_deg = deg + N;                       // N

    const long nd = (long)N * DIM;

    // out = 0, full_deg = 0
    rgcn_zero_kernel<<<4096, 256, 0, stream>>>(out, nd);
    rgcn_zero_kernel<<<256, 256, 0, stream>>>(full_deg, (long)N);

    for (int r = 0; r < NREL; ++r) {
        const int* src = edge_src + (size_t)r * E;
        const int* dst = edge_dst + (size_t)r * E;

        // ---- direction 0: ax = segsum(X[dst], src); out += ax @ w[r] + deg*bias[r]
        rgcn_zero_kernel<<<4096, 256, 0, stream>>>(AX, nd);
        rgcn_zero_kernel<<<256, 256, 0, stream>>>(deg, (long)N);
        rgcn_scatter_kernel<<<E, 128, 0, stream>>>(features, dst, src,
                                                   AX, deg, full_deg);
        rgcn_gemm_kernel<<<N / 16, 256, 0, stream>>>(AX,
                                                     w + (size_t)r * DIM * DIM,
                                                     bias + (size_t)r * DIM,
                                                     deg, out);

        // ---- direction 1: ax_t = segsum(X[src], dst); out += ax_t @ w_t[r] + deg_t*bias_t[r]
        rgcn_zero_kernel<<<4096, 256, 0, stream>>>(AX, nd);
        rgcn_zero_kernel<<<256, 256, 0, stream>>>(deg, (long)N);
        rgcn_scatter_kernel<<<E, 128, 0, stream>>>(features, src, dst,
                                                   AX, deg, full_deg);
        rgcn_gemm_kernel<<<N / 16, 256, 0, stream>>>(AX,
                                                     w_t + (size_t)r * DIM * DIM,
                                                     bias_t + (size_t)r * DIM,
                                                     deg, out);
    }

    // out *= 1 / max(full_deg, treat 0 as 1)
    rgcn_finalize_kernel<<<N, DIM / 4, 0, stream>>>(out, full_deg);
}